// SAGEConv_75101798138094
// MI455X (gfx1250) — compile-verified
//
#include <hip/hip_runtime.h>

#define NN 100000
#define NE 1250000
#define D  64

typedef float v2f __attribute__((ext_vector_type(2)));
typedef float v8f __attribute__((ext_vector_type(8)));

// ---------------------------------------------------------------------------
// Kernel 1: zero the workspace accumulators (neigh sum [N,64] + deg [N]).
// ---------------------------------------------------------------------------
__global__ void sage_zero(float* __restrict__ ws, int n) {
    int i = blockIdx.x * blockDim.x + threadIdx.x;
    if (i < n) ws[i] = 0.0f;
}

// ---------------------------------------------------------------------------
// Kernel 2: edge scatter. 16 threads per edge, one float4 chunk each.
// Consecutive threads cover one edge's contiguous 64 floats -> coalesced
// 256B source load and atomics that land in one cacheline neighborhood.
// unsafeAtomicAdd lowers to global_atomic_add_f32 (no CAS loop).
// ---------------------------------------------------------------------------
__global__ void sage_scatter(const float* __restrict__ feat,
                             const int*   __restrict__ src,
                             const int*   __restrict__ dst,
                             float*       __restrict__ neigh,
                             float*       __restrict__ deg) {
    long long idx = (long long)blockIdx.x * blockDim.x + threadIdx.x;
    const long long total = (long long)NE * 16;
    if (idx >= total) return;
    int e = (int)(idx >> 4);
    int c = (int)(idx & 15);
    int s = src[e];
    int d = dst[e];
    const float4* f4 = (const float4*)feat;
    float4 v = f4[(long long)s * 16 + c];
    float* base = neigh + (long long)d * D + c * 4;
    unsafeAtomicAdd(base + 0, v.x);
    unsafeAtomicAdd(base + 1, v.y);
    unsafeAtomicAdd(base + 2, v.z);
    unsafeAtomicAdd(base + 3, v.w);
    if (c == 0) unsafeAtomicAdd(deg + d, 1.0f);
}

// ---------------------------------------------------------------------------
// Kernel 3: fused GEMM  out = feat@Ws^T + (neigh/deg)@Wn^T + bias
// using V_WMMA_F32_16X16X4_F32 (full f32 precision).
// One wave (32 lanes) owns a 16-row node tile and produces all 64 output
// columns via 4 column-tiles. K=64 -> 16 k-steps of K=4 per stream.
//
// Fragment layouts (ISA 7.12.2, 32-bit data, wave32):
//   A 16x4:  lane L holds row M = L%16; VGPR0,1 hold K = kb+(L/16)*2 + {0,1}
//   B 4x16:  lane L holds col N = L%16; VGPR0,1 hold K = kb+(L/16)*2 + {0,1}
//            B(k,n) = W[n][k]  (row-major W gives contiguous k-pairs)
//   C/D 16x16: lane L holds col N = L%16; VGPR r holds row M = r + (L/16)*8
//
// k-loop unroll capped at 4: 32 WMMAs in flight per body (enough XDL ILP)
// while staying under 256 VGPRs (avoids s_set_vgpr_msb churn, keeps
// occupancy up to hide the strided global A loads).
// ---------------------------------------------------------------------------
__launch_bounds__(256)
__global__ void sage_gemm(const float* __restrict__ feat,
                          const float* __restrict__ Wself,
                          const float* __restrict__ Wneigh,
                          const float* __restrict__ bias,
                          const float* __restrict__ neigh,
                          const float* __restrict__ deg,
                          float*       __restrict__ out) {
    __shared__ float sWs[D * D];
    __shared__ float sWn[D * D];
    __shared__ float sB[D];

    for (int i = threadIdx.x; i < D * D; i += blockDim.x) {
        sWs[i] = Wself[i];
        sWn[i] = Wneigh[i];
    }
    if (threadIdx.x < D) sB[threadIdx.x] = bias[threadIdx.x];
    __syncthreads();

    const int lane  = threadIdx.x & 31;
    const int rc    = lane & 15;   // A-row / B-col / C-col within tile
    const int khalf = lane >> 4;   // which k-pair this half-wave holds
    const int wavesPerBlk = blockDim.x >> 5;
    const int waveId = blockIdx.x * wavesPerBlk + (threadIdx.x >> 5);
    const int nWaves = gridDim.x * wavesPerBlk;
    const int nTiles = NN / 16;    // 6250, exact

    for (int t = waveId; t < nTiles; t += nWaves) {
        const int m = t * 16 + rc;                 // this lane's A row
        // deg is integer-valued; v_rcp_f32 (~1 ulp) is plenty accurate and
        // avoids the IEEE div_scale/div_fixup expansion.
        const float inv = __builtin_amdgcn_rcpf(fmaxf(deg[m], 1.0f));

        v8f acc[4];
#pragma unroll
        for (int nt = 0; nt < 4; ++nt) {
            float bv = sB[nt * 16 + rc];           // column bias
            v8f z = {bv, bv, bv, bv, bv, bv, bv, bv};
            acc[nt] = z;
        }

        const float* arow_s = feat  + (long long)m * D;
        const float* arow_n = neigh + (long long)m * D;

#pragma unroll 4
        for (int kk = 0; kk < 16; ++kk) {
            const int kb = kk * 4 + khalf * 2;
            v2f aS = *(const v2f*)(arow_s + kb);
            v2f aN = *(const v2f*)(arow_n + kb);
            aN = aN * inv;                          // h_neigh = sum/deg
#pragma unroll
            for (int nt = 0; nt < 4; ++nt) {
                v2f bS = *(const v2f*)(&sWs[(nt * 16 + rc) * D + kb]);
                acc[nt] = __builtin_amdgcn_wmma_f32_16x16x4_f32(
                    false, aS, false, bS, (short)0, acc[nt], false, false);
            }
#pragma unroll
            for (int nt = 0; nt < 4; ++nt) {
                v2f bN = *(const v2f*)(&sWn[(nt * 16 + rc) * D + kb]);
                acc[nt] = __builtin_amdgcn_wmma_f32_16x16x4_f32(
                    false, aN, false, bN, (short)0, acc[nt], false, false);
            }
        }

        // Write back: lane holds column nt*16+rc; VGPR r is row r + khalf*8.
#pragma unroll
        for (int nt = 0; nt < 4; ++nt) {
#pragma unroll
            for (int r = 0; r < 8; ++r) {
                int mm = t * 16 + r + khalf * 8;
                out[(long long)mm * D + nt * 16 + rc] = acc[nt][r];
            }
        }
    }
}

// ---------------------------------------------------------------------------
extern "C" void kernel_launch(void* const* d_in, const int* in_sizes, int n_in,
                              void* d_out, int out_size, void* d_ws, size_t ws_size,
                              hipStream_t stream) {
    const float* feat   = (const float*)d_in[0];
    const int*   src    = (const int*)  d_in[1];
    const int*   dst    = (const int*)  d_in[2];
    const float* Wself  = (const float*)d_in[3];
    const float* Wneigh = (const float*)d_in[4];
    const float* bias   = (const float*)d_in[5];
    float*       out    = (float*)d_out;

    float* neigh = (float*)d_ws;                    // [NN*D]
    float* deg   = neigh + (size_t)NN * D;          // [NN]

    // 1) zero accumulators
    int nz = NN * D + NN;
    sage_zero<<<(nz + 255) / 256, 256, 0, stream>>>(neigh, nz);

    // 2) edge scatter (mean-aggregate numerator + degrees)
    long long total = (long long)NE * 16;
    int sblocks = (int)((total + 255) / 256);
    sage_scatter<<<sblocks, 256, 0, stream>>>(feat, src, dst, neigh, deg);

    // 3) fused WMMA GEMM + bias
    int gblocks = (NN / 16 + 7) / 8;                // one wave per 16-row tile
    sage_gemm<<<gblocks, 256, 0, stream>>>(feat, Wself, Wneigh, bias,
                                           neigh, deg, out);
}